// Dataug_85547158601809
// MI455X (gfx1250) — compile-verified
//
#include <hip/hip_runtime.h>

typedef __attribute__((ext_vector_type(2))) float v2f;
typedef __attribute__((ext_vector_type(8))) float v8f;

#define IMG_H 256
#define IMG_W 256
#define IMG_C 3
#define TILES_PER_IMG ((IMG_H / 16) * (IMG_W / 16))

// Kernel 1: MoE dispatch -> theta, folded into pixel-space coefficients.
// ixp = cx0*xs + cx1*ys + cx2 ; iyp = cy0*xs + cy1*ys + cy2
__global__ void Dataug_theta_kernel(const float* __restrict__ usmp,
                                    const float* __restrict__ nsmp,
                                    const float* __restrict__ Wt,
                                    const float* __restrict__ bt,
                                    const int* __restrict__ tsmp,
                                    const int* __restrict__ distr_flags,
                                    float* __restrict__ coef, int B) {
    int b = blockIdx.x * blockDim.x + threadIdx.x;
    if (b >= B) return;
    int e = tsmp[b];
    float s = (distr_flags[e] == 1) ? nsmp[b] : usmp[b];
    float t[6];
#pragma unroll
    for (int i = 0; i < 6; ++i) t[i] = fmaf(s, Wt[e * 6 + i], bt[e * 6 + i]);
    const float halfW = 0.5f * (float)IMG_W, halfH = 0.5f * (float)IMG_H;
    coef[b * 6 + 0] = t[0] * halfW;
    coef[b * 6 + 1] = t[1] * halfW;
    coef[b * 6 + 2] = ((t[2] + 1.0f) * (float)IMG_W - 1.0f) * 0.5f;
    coef[b * 6 + 3] = t[3] * halfH;
    coef[b * 6 + 4] = t[4] * halfH;
    coef[b * 6 + 5] = ((t[5] + 1.0f) * (float)IMG_H - 1.0f) * 0.5f;
}

// Kernel 2: one 16x16 pixel tile per wave32.
// Tile coordinates via V_WMMA_F32_16X16X4_F32, using only the K=0 row/column
// of A and B (all other K rows/cols zero -> robust to K-slot layout):
//   pass 1: D[m,n]  = ys(m) * c1            (A[m,0]=ys(m),  B[0,n]=c1)
//   pass 2: D[m,n] += 1.0   * (c0*xs(n)+c2) (A[m,0]=1.0,    B[0,n]=c0*xs(n)+c2)
// C/D layout: VGPR v, lane L -> tile row m = v + 8*(L>=16), col n = L&15.
__global__ __launch_bounds__(256) void Dataug_sample_kernel(
    const float* __restrict__ x, const float* __restrict__ coef,
    float* __restrict__ out) {
    const int lane = threadIdx.x & 31;
    const int wave = threadIdx.x >> 5;
    const int tile = blockIdx.x * 8 + wave;

    const int b = tile / TILES_PER_IMG;
    const int t = tile - b * TILES_PER_IMG;
    const int h0 = (t / (IMG_W / 16)) * 16;
    const int w0 = (t % (IMG_W / 16)) * 16;

    const int m16 = lane & 15;
    const int hihalf = lane >> 4;
    const bool lo = (hihalf == 0);

    // Wave-uniform coefficient loads (b is uniform across the wave).
    const float cx0 = coef[b * 6 + 0], cx1 = coef[b * 6 + 1], cx2 = coef[b * 6 + 2];
    const float cy0 = coef[b * 6 + 3], cy1 = coef[b * 6 + 4], cy2 = coef[b * 6 + 5];

    const float invW = 1.0f / (float)IMG_W, invH = 1.0f / (float)IMG_H;
    const float ysv = fmaf(2.0f * (float)(h0 + m16) + 1.0f, invH, -1.0f);
    const float xsv = fmaf(2.0f * (float)(w0 + m16) + 1.0f, invW, -1.0f);

    // A: 16x4, only column K=0 nonzero (lanes 0-15, first VGPR).
    v2f A1 = {lo ? ysv : 0.0f, 0.0f};
    v2f A2 = {lo ? 1.0f : 0.0f, 0.0f};
    // B: 4x16, only row K=0 nonzero (lanes 0-15, first VGPR).
    v2f Bx1 = {lo ? cx1 : 0.0f, 0.0f};
    v2f Bx2 = {lo ? fmaf(cx0, xsv, cx2) : 0.0f, 0.0f};
    v2f By1 = {lo ? cy1 : 0.0f, 0.0f};
    v2f By2 = {lo ? fmaf(cy0, xsv, cy2) : 0.0f, 0.0f};

    v8f zero = {};
    v8f dix = __builtin_amdgcn_wmma_f32_16x16x4_f32(false, A1, false, Bx1,
                                                    (short)0, zero, false, false);
    dix = __builtin_amdgcn_wmma_f32_16x16x4_f32(false, A2, false, Bx2,
                                                (short)0, dix, false, false);
    v8f diy = __builtin_amdgcn_wmma_f32_16x16x4_f32(false, A1, false, By1,
                                                    (short)0, zero, false, false);
    diy = __builtin_amdgcn_wmma_f32_16x16x4_f32(false, A2, false, By2,
                                                (short)0, diy, false, false);

    const float* img = x + (size_t)b * IMG_C * IMG_H * IMG_W;
    float* outImg = out + (size_t)b * IMG_C * IMG_H * IMG_W;

#pragma unroll
    for (int v = 0; v < 8; ++v) {
        const int h = h0 + v + 8 * hihalf;
        const int w = w0 + m16;
        const float ixp = dix[v];
        const float iyp = diy[v];

        const float fx0 = floorf(ixp), fy0 = floorf(iyp);
        const float wx1 = ixp - fx0, wy1 = iyp - fy0;
        const float wx0 = 1.0f - wx1, wy0 = 1.0f - wy1;
        const int ix0 = (int)fx0, iy0 = (int)fy0;
        const int ix1 = ix0 + 1, iy1 = iy0 + 1;

        // zero-padding validity folded into the weights
        const float vx0 = ((unsigned)ix0 < (unsigned)IMG_W) ? 1.0f : 0.0f;
        const float vx1 = ((unsigned)ix1 < (unsigned)IMG_W) ? 1.0f : 0.0f;
        const float vy0 = ((unsigned)iy0 < (unsigned)IMG_H) ? 1.0f : 0.0f;
        const float vy1 = ((unsigned)iy1 < (unsigned)IMG_H) ? 1.0f : 0.0f;

        const int xc0 = min(max(ix0, 0), IMG_W - 1);
        const int xc1 = min(max(ix1, 0), IMG_W - 1);
        const int yc0 = min(max(iy0, 0), IMG_H - 1);
        const int yc1 = min(max(iy1, 0), IMG_H - 1);

        const float w00 = wy0 * wx0 * vy0 * vx0;
        const float w01 = wy0 * wx1 * vy0 * vx1;
        const float w10 = wy1 * wx0 * vy1 * vx0;
        const float w11 = wy1 * wx1 * vy1 * vx1;

        const int o00 = yc0 * IMG_W + xc0, o01 = yc0 * IMG_W + xc1;
        const int o10 = yc1 * IMG_W + xc0, o11 = yc1 * IMG_W + xc1;

#pragma unroll
        for (int c = 0; c < IMG_C; ++c) {
            const float* p = img + c * IMG_H * IMG_W;
            float acc = p[o00] * w00;
            acc = fmaf(p[o01], w01, acc);
            acc = fmaf(p[o10], w10, acc);
            acc = fmaf(p[o11], w11, acc);
            // Streaming output (no reuse): non-temporal store keeps the
            // 201 MB output stream from evicting x's gather lines in L2.
            __builtin_nontemporal_store(acc, &outImg[(c * IMG_H + h) * IMG_W + w]);
        }
    }
}

extern "C" void kernel_launch(void* const* d_in, const int* in_sizes, int n_in,
                              void* d_out, int out_size, void* d_ws, size_t ws_size,
                              hipStream_t stream) {
    const float* x = (const float*)d_in[0];
    const float* usmp = (const float*)d_in[1];
    const float* nsmp = (const float*)d_in[2];
    const float* Wt = (const float*)d_in[3];
    const float* bt = (const float*)d_in[4];
    const int* tsmp = (const int*)d_in[5];
    const int* distr_flags = (const int*)d_in[6];
    float* out = (float*)d_out;
    float* coef = (float*)d_ws;  // B*6 floats

    const int B = in_sizes[1];  // usmp is [B,1]

    Dataug_theta_kernel<<<(B + 255) / 256, 256, 0, stream>>>(
        usmp, nsmp, Wt, bt, tsmp, distr_flags, coef, B);

    const int tiles = B * TILES_PER_IMG;  // one tile per wave32
    Dataug_sample_kernel<<<tiles / 8, 256, 0, stream>>>(x, coef, out);
}